// GAT_Network_13417477833474
// MI455X (gfx1250) — compile-verified
//
#include <hip/hip_runtime.h>
#include <cstdint>

typedef __attribute__((ext_vector_type(2))) float v2f;
typedef __attribute__((ext_vector_type(8))) float v8f;

#define NEG_SLOPE 0.2f
#define GAT_EPS 1e-16f

__device__ __forceinline__ void atomicMaxF(float* addr, float v) {
  // Bit-pun float max: positive floats order as ints, negative floats order
  // reversed as uints. Mixed updates to the same word remain correct.
  if (v >= 0.0f) atomicMax((int*)addr, __float_as_int(v));
  else           atomicMin((unsigned int*)addr, __float_as_uint(v));
}

// ---------------------------------------------------------------------------
// Dual projection GEMM: Cl = A @ Bl, Cr = A @ Br via V_WMMA_F32_16X16X4_F32.
// A:[M,K] row-major, B*:[K,Ncol] row-major, C*:[M,Ncol] row-major, fp32.
// Block = 4 waves. The block's 16-row A tile spans the full K, i.e. it is a
// contiguous 16*K*4B region: stage it once in LDS (<=32KB), share across the
// 4 waves (A-fragments come from ds_load_b64). Each wave owns one 16x16
// output tile of BOTH Cl and Cr (A-fragment reused across the two WMMAs).
// Per ISA 7.12.2 (fp32 16x16x4): lane r=M; half=lane>>4 selects K {0,2}/{1,3}
// across the two A VGPRs; B mirrors with lane=N; C: VGPR i -> M=i+8*half.
// ---------------------------------------------------------------------------
__global__ __launch_bounds__(128) void gemm_dual_wmma(
    const float* __restrict__ A, const float* __restrict__ Bl,
    const float* __restrict__ Br, float* __restrict__ Cl,
    float* __restrict__ Cr, int K, int Ncol) {
  __shared__ __align__(16) float As[16 * 512];   // max K = 512 -> 32 KB

  {  // cooperative contiguous copy of the 16 x K A tile
    const float4* Ag  = (const float4*)(A + (size_t)blockIdx.x * 16 * K);
    float4*       As4 = (float4*)As;
    const int n4 = 4 * K;                        // 16*K/4 float4 elements
    for (int i = threadIdx.x; i < n4; i += 128) As4[i] = Ag[i];
  }
  __syncthreads();

  const int lane   = threadIdx.x & 31;
  const int wave   = threadIdx.x >> 5;
  const int nTile  = blockIdx.y * 4 + wave;
  const bool active = (nTile * 16 < Ncol);       // wave-uniform
  const int n0     = active ? nTile * 16 : 0;    // clamp: idle waves duplicate
  const int half   = lane >> 4;
  const int r      = lane & 15;

  const float* as  = As + (size_t)r * K + 2 * half;
  const float* blp = Bl + (size_t)(2 * half) * Ncol + n0 + r;
  const float* brp = Br + (size_t)(2 * half) * Ncol + n0 + r;

  v8f accl = {}; v8f accr = {};
  for (int k0 = 0; k0 < K; k0 += 4) {
    v2f a, bl, br;
    a.x  = as[0];   a.y  = as[1];          // LDS: A[m0+r][k0+2h], [k0+2h+1]
    bl.x = blp[0];  bl.y = blp[Ncol];      // Bl[k0+2h][n0+r], Bl[k0+2h+1][n0+r]
    br.x = brp[0];  br.y = brp[Ncol];
    accl = __builtin_amdgcn_wmma_f32_16x16x4_f32(false, a, false, bl,
                                                 (short)0, accl, false, false);
    accr = __builtin_amdgcn_wmma_f32_16x16x4_f32(false, a, false, br,
                                                 (short)0, accr, false, false);
    as  += 4;
    blp += (size_t)4 * Ncol;
    brp += (size_t)4 * Ncol;
  }
  if (active) {
    const int m0 = blockIdx.x * 16;
    float* cl = Cl + (size_t)(m0 + 8 * half) * Ncol + n0 + r;
    float* cr = Cr + (size_t)(m0 + 8 * half) * Ncol + n0 + r;
#pragma unroll
    for (int i = 0; i < 8; ++i) {
      cl[(size_t)i * Ncol] = accl[i];
      cr[(size_t)i * Ncol] = accr[i];
    }
  }
}

// ---------------------------------------------------------------------------
// Edge-phase kernels (segment softmax + weighted scatter). dtype int64 edges.
// Edge id e < E: src=ei[e], dst=ei[E+e]; e >= E: self-loop (src=dst=e-E).
// ---------------------------------------------------------------------------
__global__ void k_init_m_denom(float* __restrict__ m, float* __restrict__ d,
                               long long n) {
  long long t = (long long)blockIdx.x * blockDim.x + threadIdx.x;
  if (t >= n) return;
  m[t] = -__builtin_inff();
  d[t] = 0.0f;
}

// One group of G = C/4 lanes per (edge, head): fully coalesced float4 gathers
// of xl[src]/xr[dst], LeakyReLU-dot with att, shfl-tree reduce within group.
// G in {16,8,4} divides both 32 (wave) and 256 (block), so groups never
// straddle waves and tail masking drops whole groups uniformly.
__global__ void k_edge_logits(const long long* __restrict__ ei, int E, int N,
                              int H, int C, const float* __restrict__ xl,
                              const float* __restrict__ xr,
                              const float* __restrict__ att,
                              float* __restrict__ logit, float* __restrict__ m) {
  const int G = C >> 2;
  long long t = (long long)blockIdx.x * blockDim.x + threadIdx.x;
  long long g = t / G;
  const int lane_in = threadIdx.x & (G - 1);     // G divides blockDim
  long long totalg = (long long)(E + N) * H;
  if (g >= totalg) return;
  int e = (int)(g / H);
  int h = (int)(g - (long long)e * H);
  long long s, d;
  if (e < E) { s = ei[e]; d = ei[(size_t)E + e]; }
  else       { s = d = e - E; }
  const int HC = H * C;
  const int c  = lane_in << 2;
  const float4 l  = *(const float4*)(xl  + (size_t)s * HC + h * C + c);
  const float4 rr = *(const float4*)(xr  + (size_t)d * HC + h * C + c);
  const float4 a  = *(const float4*)(att + (size_t)h * C + c);
  float v0 = l.x + rr.x; v0 = v0 > 0.f ? v0 : NEG_SLOPE * v0;
  float v1 = l.y + rr.y; v1 = v1 > 0.f ? v1 : NEG_SLOPE * v1;
  float v2 = l.z + rr.z; v2 = v2 > 0.f ? v2 : NEG_SLOPE * v2;
  float v3 = l.w + rr.w; v3 = v3 > 0.f ? v3 : NEG_SLOPE * v3;
  float acc = a.x * v0 + a.y * v1 + a.z * v2 + a.w * v3;
#pragma unroll
  for (int off = 16; off > 0; off >>= 1)
    if (off < G) acc += __shfl_down(acc, off, G);
  if (lane_in == 0) {
    logit[g] = acc;
    atomicMaxF(&m[(size_t)d * H + h], acc);      // segment_max over dst
  }
}

__global__ void k_edge_exp(const long long* __restrict__ ei, int E, int N,
                           int H, const float* __restrict__ m,
                           float* __restrict__ logit, float* __restrict__ denom) {
  long long t = (long long)blockIdx.x * blockDim.x + threadIdx.x;
  long long total = (long long)(E + N) * H;
  if (t >= total) return;
  int e = (int)(t / H);
  int h = (int)(t - (long long)e * H);
  long long d = (e < E) ? ei[(size_t)E + e] : (long long)(e - E);
  float u = expf(logit[t] - m[(size_t)d * H + h]);
  logit[t] = u;                              // overwrite with unnormalized alpha
  atomicAdd(&denom[(size_t)d * H + h], u);   // segment_sum over dst
}

__global__ void k_init_bias4(float4* __restrict__ out,
                             const float* __restrict__ bias, long long n4,
                             int HC4) {
  long long t = (long long)blockIdx.x * blockDim.x + threadIdx.x;
  if (t >= n4) return;
  int j = (int)(t % HC4) << 2;
  float4 v; v.x = bias[j]; v.y = bias[j + 1]; v.z = bias[j + 2]; v.w = bias[j + 3];
  out[t] = v;
}

// One thread per (edge, 4-channel chunk): coalesced float4 load of xl[src],
// alpha scale, 4 fp32 global atomics into out[dst].
__global__ void k_edge_aggr(const long long* __restrict__ ei, int E, int N,
                            int H, int C, const float* __restrict__ xl,
                            const float* __restrict__ logit,
                            const float* __restrict__ denom,
                            float* __restrict__ out) {
  const int HC  = H * C;
  const int HC4 = HC >> 2;
  long long t = (long long)blockIdx.x * blockDim.x + threadIdx.x;
  long long total = (long long)(E + N) * HC4;
  if (t >= total) return;
  int e = (int)(t / HC4);
  int j = (int)(t - (long long)e * HC4) << 2;
  int h = j / C;
  long long s, d;
  if (e < E) { s = ei[e]; d = ei[(size_t)E + e]; }
  else       { s = d = e - E; }
  float alpha = logit[(size_t)e * H + h] / (denom[(size_t)d * H + h] + GAT_EPS);
  float4 v = *(const float4*)(xl + (size_t)s * HC + j);
  float* po = out + (size_t)d * HC + j;
  atomicAdd(po + 0, alpha * v.x);
  atomicAdd(po + 1, alpha * v.y);
  atomicAdd(po + 2, alpha * v.z);
  atomicAdd(po + 3, alpha * v.w);
}

__global__ void k_relu4(float4* __restrict__ p, long long n4) {
  long long t = (long long)blockIdx.x * blockDim.x + threadIdx.x;
  if (t >= n4) return;
  float4 v = p[t];
  v.x = v.x > 0.f ? v.x : 0.f;
  v.y = v.y > 0.f ? v.y : 0.f;
  v.z = v.z > 0.f ? v.z : 0.f;
  v.w = v.w > 0.f ? v.w : 0.f;
  p[t] = v;
}

// Final MLP head: out = relu(emb @ fc1_w + fc1_b) @ fc2_w + fc2_b
__global__ void k_fc(const float* __restrict__ emb,
                     const float* __restrict__ w1, const float* __restrict__ b1,
                     const float* __restrict__ w2, const float* __restrict__ b2,
                     float* __restrict__ out, int N) {
  int n = blockIdx.x * blockDim.x + threadIdx.x;
  if (n >= N) return;
  float x[16];
  const float4* p = (const float4*)(emb + (size_t)n * 16);
#pragma unroll
  for (int q = 0; q < 4; ++q) {
    float4 v = p[q];
    x[4 * q] = v.x; x[4 * q + 1] = v.y; x[4 * q + 2] = v.z; x[4 * q + 3] = v.w;
  }
  float hbuf[8];
#pragma unroll
  for (int j = 0; j < 8; ++j) {
    float h = b1[j];
#pragma unroll
    for (int i = 0; i < 16; ++i) h += x[i] * w1[i * 8 + j];
    hbuf[j] = h > 0.0f ? h : 0.0f;
  }
#pragma unroll
  for (int k = 0; k < 2; ++k) {
    float o = b2[k];
#pragma unroll
    for (int j = 0; j < 8; ++j) o += hbuf[j] * w2[j * 2 + k];
    out[(size_t)n * 2 + k] = o;
  }
}

// ---------------------------------------------------------------------------
extern "C" void kernel_launch(void* const* d_in, const int* in_sizes, int n_in,
                              void* d_out, int out_size, void* d_ws, size_t ws_size,
                              hipStream_t stream) {
  (void)n_in; (void)out_size; (void)ws_size;
  const float*     x    = (const float*)d_in[0];
  const long long* ei   = (const long long*)d_in[1];   // int64 edge_index [2,E]
  const float* Wl1 = (const float*)d_in[2];
  const float* Wr1 = (const float*)d_in[3];
  const float* att1= (const float*)d_in[4];
  const float* b1  = (const float*)d_in[5];
  const float* Wl2 = (const float*)d_in[6];
  const float* Wr2 = (const float*)d_in[7];
  const float* att2= (const float*)d_in[8];
  const float* b2  = (const float*)d_in[9];
  const float* Wl3 = (const float*)d_in[10];
  const float* Wr3 = (const float*)d_in[11];
  const float* att3= (const float*)d_in[12];
  const float* b3  = (const float*)d_in[13];
  const float* fc1w= (const float*)d_in[14];
  const float* fc1b= (const float*)d_in[15];
  const float* fc2w= (const float*)d_in[16];
  const float* fc2b= (const float*)d_in[17];

  const int N = in_sizes[0] / 256;     // 50000 (F_in = 256)
  const int E = in_sizes[1] / 2;       // 800000

  // Workspace layout (floats)
  float* ws = (float*)d_ws;
  size_t o = 0;
  float* xl = ws + o;  o += (size_t)N * 512;
  float* xr = ws + o;  o += (size_t)N * 512;
  float* hA = ws + o;  o += (size_t)N * 512;   // layer-1 output
  float* hB = ws + o;  o += (size_t)N * 128;   // layer-2 output
  float* lg = ws + o;  o += (size_t)(E + N) * 8;
  float* mb = ws + o;  o += (size_t)N * 8;
  float* db = ws + o;

  float* outp = (float*)d_out;
  float* emb  = outp + (size_t)N * 2;          // embeddings slice [N,16]

  auto layer = [&](const float* xin, int K, const float* Wl, const float* Wr,
                   const float* att, const float* bias, int H, int C,
                   float* hout) {
    const int HC = H * C;
    dim3 ggrid(N / 16, (HC / 16 + 3) / 4);
    gemm_dual_wmma<<<ggrid, 128, 0, stream>>>(xin, Wl, Wr, xl, xr, K, HC);

    long long nh = (long long)N * H;
    k_init_m_denom<<<(unsigned)((nh + 255) / 256), 256, 0, stream>>>(mb, db, nh);

    long long lgt = (long long)(E + N) * H * (C >> 2);   // group-per-(e,h)
    k_edge_logits<<<(unsigned)((lgt + 255) / 256), 256, 0, stream>>>(
        ei, E, N, H, C, xl, xr, att, lg, mb);

    long long eh = (long long)(E + N) * H;
    k_edge_exp<<<(unsigned)((eh + 255) / 256), 256, 0, stream>>>(
        ei, E, N, H, lg, mb, db);

    long long nhc4 = (long long)N * (HC >> 2);
    k_init_bias4<<<(unsigned)((nhc4 + 255) / 256), 256, 0, stream>>>(
        (float4*)hout, bias, nhc4, HC >> 2);

    long long ehc4 = (long long)(E + N) * (HC >> 2);
    k_edge_aggr<<<(unsigned)((ehc4 + 255) / 256), 256, 0, stream>>>(
        ei, E, N, H, C, xl, lg, db, hout);

    k_relu4<<<(unsigned)((nhc4 + 255) / 256), 256, 0, stream>>>(
        (float4*)hout, nhc4);
  };

  layer(x,  256, Wl1, Wr1, att1, b1, 8, 64, hA);   // 256 -> 8x64 = 512
  layer(hA, 512, Wl2, Wr2, att2, b2, 4, 32, hB);   // 512 -> 4x32 = 128
  layer(hB, 128, Wl3, Wr3, att3, b3, 1, 16, emb);  // 128 -> 1x16 = 16

  k_fc<<<(N + 255) / 256, 256, 0, stream>>>(emb, fc1w, fc1b, fc2w, fc2b, outp, N);
}